// SACN_29721173688344
// MI455X (gfx1250) — compile-verified
//
#include <hip/hip_runtime.h>

// ---------------------------------------------------------------------------
// SACN pipeline for MI455X (gfx1250, wave32).
// All dense GEMMs run on the matrix pipe via V_WMMA_F32_16X16X4_F32, one wave
// per 16x64 output strip (4 accumulators, A fragment reused 4x). Hot loop is
// branch-free and software-pipelined (next K-step fragments load while the
// current WMMAs issue); B stride is a template constant so k-advance folds
// into immediate offsets. Graph scatter uses global_atomic_add_f32.
// ---------------------------------------------------------------------------

typedef __attribute__((ext_vector_type(2))) float v2f;
typedef __attribute__((ext_vector_type(8))) float v8f;

namespace {
constexpr int   kN    = 100000;  // nodes
constexpr int   kINIT = 100;
constexpr int   kGC1  = 150;
constexpr int   kD    = 200;
constexpr int   kC    = 200;     // conv out channels
constexpr int   kKw   = 5;       // conv kernel width
constexpr int   kB    = 128;     // batch
constexpr int   kNA   = 8;
constexpr int   kE    = 500000;  // edges
constexpr int   kNC   = 16;
constexpr float kRS   = 0.99999500003749937f;  // 1/sqrt(1 + 1e-5)

enum Act { ACT_NONE = 0, ACT_RELU = 1, ACT_LRELU = 2, ACT_SIGMOID = 3 };
}  // namespace

// --------------------------- utility kernels -------------------------------

__global__ void fill_zero(float* __restrict__ p, int n) {
  int i = blockIdx.x * blockDim.x + threadIdx.x;
  const int stride = gridDim.x * blockDim.x;
  for (; i < n; i += stride) p[i] = 0.0f;
}

// out[row] += w*support[col]; out[col] += w*support[row]  (per feature f)
__global__ void gcn_scatter(const float* __restrict__ support,
                            float* __restrict__ out,
                            const int* __restrict__ edge_index,
                            const int* __restrict__ edge_rel,
                            const float* __restrict__ alpha, int E, int F) {
  long long t = (long long)blockIdx.x * blockDim.x + threadIdx.x;
  const long long total = (long long)E * F;
  if (t >= total) return;
  const int e = (int)(t / F);
  const int f = (int)(t - (long long)e * F);
  const int r = edge_index[e];
  const int c = edge_index[E + e];
  const float w = alpha[edge_rel[e]];  // alpha shape (R+1,1)
  atomicAdd(&out[(long long)r * F + f], w * support[(long long)c * F + f]);
  atomicAdd(&out[(long long)c * F + f], w * support[(long long)r * F + f]);
}

// x = tanh( (x + bias[f]) * (g[f]*kRS) + b[f] )   in place
__global__ void bias_bn_tanh(float* __restrict__ x, const float* __restrict__ bias,
                             const float* __restrict__ g, const float* __restrict__ b,
                             int n, int F) {
  int i = blockIdx.x * blockDim.x + threadIdx.x;
  if (i >= n) return;
  const int f = i % F;
  float v = x[i] + bias[f];
  v = v * (g[f] * kRS) + b[f];
  x[i] = tanhf(v);
}

// s[b,0,:] = all_emb[e1[b]]; s[b,1,:] = emb_rel[rel[b]]; then per-channel bn0
__global__ void build_s(const float* __restrict__ all_emb,
                        const float* __restrict__ emb_rel,
                        const int* __restrict__ e1, const int* __restrict__ rel,
                        const float* __restrict__ g0, const float* __restrict__ b0,
                        float* __restrict__ s) {
  int i = blockIdx.x * blockDim.x + threadIdx.x;
  if (i >= kB * 2 * kD) return;
  const int d = i % kD;
  const int ch = (i / kD) & 1;
  const int b = i / (2 * kD);
  float v = (ch == 0) ? all_emb[(long long)e1[b] * kD + d]
                      : emb_rel[(long long)rel[b] * kD + d];
  s[i] = v * (g0[ch] * kRS) + b0[ch];
}

// 1D conv (pad K/2) + conv bias + bn1 + relu
__global__ void conv_bn_relu(const float* __restrict__ s, const float* __restrict__ w,
                             const float* __restrict__ cb, const float* __restrict__ g,
                             const float* __restrict__ bb, float* __restrict__ out) {
  int i = blockIdx.x * blockDim.x + threadIdx.x;
  if (i >= kB * kC * kD) return;
  const int d = i % kD;
  const int c = (i / kD) % kC;
  const int b = i / (kC * kD);
  float acc = 0.0f;
#pragma unroll
  for (int ch = 0; ch < 2; ++ch) {
#pragma unroll
    for (int k = 0; k < kKw; ++k) {
      const int dd = d + k - kKw / 2;
      if (dd < 0 || dd >= kD) continue;
      acc += s[(b * 2 + ch) * kD + dd] * w[(c * 2 + ch) * kKw + k];
    }
  }
  acc += cb[c];
  acc = acc * (g[c] * kRS) + bb[c];
  out[i] = fmaxf(acc, 0.0f);
}

// ssum = sum_a all_emb[attr[b,a]]; deep = 0.5*(ssum^2 - sum_a v^2)
__global__ void nfm_prep(const float* __restrict__ all_emb,
                         const int* __restrict__ attr,
                         float* __restrict__ ssum, float* __restrict__ deep) {
  int i = blockIdx.x * blockDim.x + threadIdx.x;
  if (i >= kB * kD) return;
  const int d = i % kD;
  const int b = i / kD;
  float s = 0.0f, sq = 0.0f;
#pragma unroll
  for (int a = 0; a < kNA; ++a) {
    const float v = all_emb[(long long)attr[b * kNA + a] * kD + d];
    s += v;
    sq += v * v;
  }
  ssum[i] = s;
  deep[i] = 0.5f * (s * s - sq);
}

// cc[:, :D] = nfm1 + nfm2 ; cc[:, D:2D] = xfc
__global__ void concat_nfm(const float* __restrict__ n1, const float* __restrict__ n2,
                           const float* __restrict__ xfc, float* __restrict__ cc) {
  int i = blockIdx.x * blockDim.x + threadIdx.x;
  if (i >= kB * 2 * kD) return;
  const int d = i % (2 * kD);
  const int b = i / (2 * kD);
  cc[i] = (d < kD) ? (n1[b * kD + d] + n2[b * kD + d]) : xfc[b * kD + (d - kD)];
}

// --------------------------- WMMA f32 GEMM ---------------------------------
// C(MxN) = act( bn( A(MxK) * B + bias ) )
// TRANSB=false: B is KxN row-major (B[k*LDB+n]).
// TRANSB=true : B is NxK row-major (B[n*LDB+k])  i.e.  C = A * B^T.
// LDB is a compile-time stride so k-advance folds into immediate offsets.
// One wave computes a 16 x (16*NT) strip: NT accumulators share one A fragment
// per K-step. K stepped by 4 via V_WMMA_F32_16X16X4_F32, software-pipelined.
// Lane layout per ISA 7.12.2:
//   A 16x4: lanes 0-15 hold row m, K={k,k+1}; lanes 16-31 hold K={k+2,k+3}.
//   B 4x16: vgpr0 = rows {k,k+2} split across lane halves, vgpr1 = {k+1,k+3}.
//   C/D:    vgpr i -> row m0+i (lanes 0-15) / m0+i+8 (lanes 16-31), col = lane%16.
// Out-of-range M/N lanes: addresses are clamped (loads stay in-bounds); their
// products only affect rows/cols we never store. Only the K tail zeroes values.
template <bool TRANSB, int NT, int LDB>
__global__ void __launch_bounds__(256)
wmma_gemm_f32(const float* __restrict__ A, const float* __restrict__ Bm,
              float* __restrict__ Cm, int M, int N, int K,
              int lda, int ldc,
              const float* __restrict__ bias,
              const float* __restrict__ bng,
              const float* __restrict__ bnb, int act) {
  const int lane = threadIdx.x & 31;
  const int wave = threadIdx.x >> 5;
  const int stripsN = (N + 16 * NT - 1) / (16 * NT);
  const int tilesM = (M + 15) >> 4;
  const int strip = blockIdx.x * (blockDim.x >> 5) + wave;
  if (strip >= tilesM * stripsN) return;  // wave-uniform: EXEC stays all-ones
  const int tm = strip / stripsN;
  const int sn = strip - tm * stripsN;
  const int m0 = tm << 4;
  const int n0 = sn * 16 * NT;
  const int half = lane >> 4;   // which K pair this lane carries
  const int l15 = lane & 15;

  const int arow = min(m0 + l15, M - 1);          // clamped, always in-bounds
  int bcol[NT];
#pragma unroll
  for (int t = 0; t < NT; ++t) bcol[t] = min(n0 + 16 * t + l15, N - 1);

  v8f acc[NT];
#pragma unroll
  for (int t = 0; t < NT; ++t) acc[t] = (v8f){};

  const int K4 = K & ~3;

  // fragment pointers (even float offsets -> v2f loads are 8B aligned)
  const float* aptr = A + (long long)arow * lda + 2 * half;
  const float* bptr[NT];
#pragma unroll
  for (int t = 0; t < NT; ++t)
    bptr[t] = TRANSB ? (Bm + (long long)bcol[t] * LDB + 2 * half)
                     : (Bm + (long long)(2 * half) * LDB + bcol[t]);

  auto load_b = [&](int t) -> v2f {
    v2f b;
    if (TRANSB) {
      b = *(const v2f*)bptr[t];
    } else {
      b.x = bptr[t][0];
      b.y = bptr[t][LDB];
    }
    return b;
  };

  if (K4 >= 4) {  // software-pipelined main loop
    v2f a_cur = *(const v2f*)aptr;
    v2f b_cur[NT];
#pragma unroll
    for (int t = 0; t < NT; ++t) b_cur[t] = load_b(t);

    for (int k = 4; k < K4; k += 4) {
      aptr += 4;
#pragma unroll
      for (int t = 0; t < NT; ++t) bptr[t] += TRANSB ? 4 : 4 * LDB;
      const v2f a_nxt = *(const v2f*)aptr;   // prefetch next K-step fragments
      v2f b_nxt[NT];
#pragma unroll
      for (int t = 0; t < NT; ++t) b_nxt[t] = load_b(t);
#pragma unroll
      for (int t = 0; t < NT; ++t)
        acc[t] = __builtin_amdgcn_wmma_f32_16x16x4_f32(false, a_cur, false, b_cur[t],
                                                       (short)0, acc[t], false, false);
      a_cur = a_nxt;
#pragma unroll
      for (int t = 0; t < NT; ++t) b_cur[t] = b_nxt[t];
    }
#pragma unroll
    for (int t = 0; t < NT; ++t)
      acc[t] = __builtin_amdgcn_wmma_f32_16x16x4_f32(false, a_cur, false, b_cur[t],
                                                     (short)0, acc[t], false, false);
  }

  if (K4 < K) {  // K tail: unconditional clamped loads + value selects
    const int k0 = K4 + 2 * half;
    const bool ok0 = (k0 + 0) < K;
    const bool ok1 = (k0 + 1) < K;
    const int kc0 = min(k0 + 0, K - 1);
    const int kc1 = min(k0 + 1, K - 1);
    const float av0 = A[(long long)arow * lda + kc0];
    const float av1 = A[(long long)arow * lda + kc1];
    v2f a;
    a.x = ok0 ? av0 : 0.0f;
    a.y = ok1 ? av1 : 0.0f;
#pragma unroll
    for (int t = 0; t < NT; ++t) {
      float bv0, bv1;
      if (TRANSB) {
        bv0 = Bm[(long long)bcol[t] * LDB + kc0];
        bv1 = Bm[(long long)bcol[t] * LDB + kc1];
      } else {
        bv0 = Bm[(long long)kc0 * LDB + bcol[t]];
        bv1 = Bm[(long long)kc1 * LDB + bcol[t]];
      }
      v2f b;
      b.x = ok0 ? bv0 : 0.0f;
      b.y = ok1 ? bv1 : 0.0f;
      acc[t] = __builtin_amdgcn_wmma_f32_16x16x4_f32(false, a, false, b, (short)0,
                                                     acc[t], false, false);
    }
  }

  // epilogue + store (divergence only after the last WMMA)
#pragma unroll
  for (int t = 0; t < NT; ++t) {
    const int col = n0 + 16 * t + l15;
    if (col >= N) continue;
    const float bv = bias ? bias[col] : 0.0f;
    const float gv = bng ? bng[col] * kRS : 1.0f;
    const float bbv = bnb ? bnb[col] : 0.0f;
#pragma unroll
    for (int i = 0; i < 8; ++i) {
      const int row = m0 + i + 8 * half;
      if (row >= M) continue;
      float v = acc[t][i] + bv;
      if (bng) v = v * gv + bbv;
      if (act == ACT_RELU) v = fmaxf(v, 0.0f);
      else if (act == ACT_LRELU) v = (v >= 0.0f) ? v : 0.01f * v;
      else if (act == ACT_SIGMOID) v = 1.0f / (1.0f + __expf(-v));
      Cm[(long long)row * ldc + col] = v;
    }
  }
}

// ------------------------------ launcher -----------------------------------

static inline int ceil_div_i(long long a, long long b) { return (int)((a + b - 1) / b); }

extern "C" void kernel_launch(void* const* d_in, const int* in_sizes, int n_in,
                              void* d_out, int out_size, void* d_ws, size_t ws_size,
                              hipStream_t stream) {
  (void)in_sizes; (void)n_in; (void)out_size; (void)ws_size;

  const int*   e1         = (const int*)d_in[0];
  const int*   rel        = (const int*)d_in[1];
  const int*   attr       = (const int*)d_in[2];
  /* d_in[3] = X = arange(N): identity gather, unused */
  const int*   edge_index = (const int*)d_in[4];
  const int*   edge_rel   = (const int*)d_in[5];
  const float* emb_e      = (const float*)d_in[6];
  const float* emb_rel    = (const float*)d_in[7];
  const float* alpha1     = (const float*)d_in[8];
  const float* alpha2     = (const float*)d_in[9];
  const float* gc1_W      = (const float*)d_in[10];
  const float* gc1_b      = (const float*)d_in[11];
  const float* gc2_W      = (const float*)d_in[12];
  const float* gc2_b      = (const float*)d_in[13];
  const float* bn3_g      = (const float*)d_in[14];
  const float* bn3_b      = (const float*)d_in[15];
  const float* bn4_g      = (const float*)d_in[16];
  const float* bn4_b      = (const float*)d_in[17];
  const float* bn0_g      = (const float*)d_in[18];
  const float* bn0_b      = (const float*)d_in[19];
  const float* conv_w     = (const float*)d_in[20];
  const float* conv_b     = (const float*)d_in[21];
  const float* bn1_g      = (const float*)d_in[22];
  const float* bn1_b      = (const float*)d_in[23];
  const float* fc_w       = (const float*)d_in[24];
  const float* fc_b       = (const float*)d_in[25];
  const float* bn2_g      = (const float*)d_in[26];
  const float* bn2_b      = (const float*)d_in[27];
  const float* bi_w       = (const float*)d_in[28];
  const float* bi_b       = (const float*)d_in[29];
  const float* si_w       = (const float*)d_in[30];
  const float* si_b       = (const float*)d_in[31];
  const float* cs_w       = (const float*)d_in[32];
  const float* cs_b       = (const float*)d_in[33];
  const float* fc2_w      = (const float*)d_in[34];
  const float* fc2_b      = (const float*)d_in[35];

  // workspace layout (floats)
  float* ws    = (float*)d_ws;
  float* bufA  = ws;                                 // N*D  (GEMM outputs)
  float* bufB  = bufA + (size_t)kN * kD;             // N*D  (scatter acc / all_emb)
  float* s_bn  = bufB + (size_t)kN * kD;             // B*2*D
  float* xconv = s_bn + (size_t)kB * 2 * kD;         // B*C*D
  float* xfc   = xconv + (size_t)kB * kC * kD;       // B*D
  float* ssum  = xfc  + (size_t)kB * kD;             // B*D
  float* deep  = ssum + (size_t)kB * kD;             // B*D
  float* nfm1  = deep + (size_t)kB * kD;             // B*D
  float* nfm2  = nfm1 + (size_t)kB * kD;             // B*D
  float* ccat  = nfm2 + (size_t)kB * kD;             // B*2*D
  float* user  = ccat + (size_t)kB * 2 * kD;         // B*D

  float* pred     = (float*)d_out;                   // B*N
  float* clusters = pred + (size_t)kB * kN;          // B*NC

  const int TPB = 256;  // 8 waves per block
  const int WPB = TPB / 32;
  constexpr int NT = 4;  // 16x64 strip per wave
  auto gemm_blocks = [&](int M, int N, int nt) {
    const long long strips =
        (long long)((M + 15) / 16) * ((N + 16 * nt - 1) / (16 * nt));
    return ceil_div_i(strips, WPB);
  };

  // ---- GCN layer 1 --------------------------------------------------------
  wmma_gemm_f32<false, NT, kGC1><<<gemm_blocks(kN, kGC1, NT), TPB, 0, stream>>>(
      emb_e, gc1_W, bufA, kN, kGC1, kINIT, kINIT, kGC1,
      nullptr, nullptr, nullptr, ACT_NONE);
  fill_zero<<<2048, TPB, 0, stream>>>(bufB, kN * kGC1);
  gcn_scatter<<<ceil_div_i((long long)kE * kGC1, TPB), TPB, 0, stream>>>(
      bufA, bufB, edge_index, edge_rel, alpha1, kE, kGC1);
  bias_bn_tanh<<<ceil_div_i((long long)kN * kGC1, TPB), TPB, 0, stream>>>(
      bufB, gc1_b, bn3_g, bn3_b, kN * kGC1, kGC1);

  // ---- GCN layer 2 --------------------------------------------------------
  wmma_gemm_f32<false, NT, kD><<<gemm_blocks(kN, kD, NT), TPB, 0, stream>>>(
      bufB, gc2_W, bufA, kN, kD, kGC1, kGC1, kD,
      nullptr, nullptr, nullptr, ACT_NONE);
  fill_zero<<<2048, TPB, 0, stream>>>(bufB, kN * kD);
  gcn_scatter<<<ceil_div_i((long long)kE * kD, TPB), TPB, 0, stream>>>(
      bufA, bufB, edge_index, edge_rel, alpha2, kE, kD);
  bias_bn_tanh<<<ceil_div_i((long long)kN * kD, TPB), TPB, 0, stream>>>(
      bufB, gc2_b, bn4_g, bn4_b, kN * kD, kD);   // bufB = all_emb

  // ---- ConvE head ---------------------------------------------------------
  build_s<<<ceil_div_i(kB * 2 * kD, TPB), TPB, 0, stream>>>(
      bufB, emb_rel, e1, rel, bn0_g, bn0_b, s_bn);
  conv_bn_relu<<<ceil_div_i((long long)kB * kC * kD, TPB), TPB, 0, stream>>>(
      s_bn, conv_w, conv_b, bn1_g, bn1_b, xconv);
  wmma_gemm_f32<true, NT, kC * kD><<<gemm_blocks(kB, kD, NT), TPB, 0, stream>>>(
      xconv, fc_w, xfc, kB, kD, kC * kD, kC * kD, kD,
      fc_b, bn2_g, bn2_b, ACT_RELU);

  // ---- NFM block ----------------------------------------------------------
  nfm_prep<<<ceil_div_i(kB * kD, TPB), TPB, 0, stream>>>(bufB, attr, ssum, deep);
  wmma_gemm_f32<true, NT, kD><<<gemm_blocks(kB, kD, NT), TPB, 0, stream>>>(
      deep, bi_w, nfm1, kB, kD, kD, kD, kD, bi_b, nullptr, nullptr, ACT_LRELU);
  wmma_gemm_f32<true, NT, kD><<<gemm_blocks(kB, kD, NT), TPB, 0, stream>>>(
      ssum, si_w, nfm2, kB, kD, kD, kD, kD, si_b, nullptr, nullptr, ACT_LRELU);
  concat_nfm<<<ceil_div_i(kB * 2 * kD, TPB), TPB, 0, stream>>>(nfm1, nfm2, xfc, ccat);
  wmma_gemm_f32<true, NT, 2 * kD><<<gemm_blocks(kB, kD, NT), TPB, 0, stream>>>(
      ccat, cs_w, user, kB, kD, 2 * kD, 2 * kD, kD,
      cs_b, nullptr, nullptr, ACT_LRELU);

  // ---- outputs ------------------------------------------------------------
  wmma_gemm_f32<true, NT, kD><<<gemm_blocks(kB, kN, NT), TPB, 0, stream>>>(
      user, bufB, pred, kB, kN, kD, kD, kN,
      nullptr, nullptr, nullptr, ACT_SIGMOID);
  wmma_gemm_f32<true, 1, kD><<<gemm_blocks(kB, kNC, 1), TPB, 0, stream>>>(
      user, fc2_w, clusters, kB, kNC, kD, kD, kNC,
      fc2_b, nullptr, nullptr, ACT_SIGMOID);
}